// VitsEncoder_11759620456870
// MI455X (gfx1250) — compile-verified
//
#include <hip/hip_runtime.h>

typedef __attribute__((ext_vector_type(2))) float v2f;
typedef __attribute__((ext_vector_type(8))) float v8f;

#define TT   16384   // input time
#define TP   8192    // conv output time (stride 2)
#define DD   768     // channels
#define K1   1536    // conv GEMM K (= D * kernel_size)
#define NBINS 1024
#define LKP  20      // padded LDS row stride (dwords): 80B rows -> b128-aligned stores,
                     // 8B-aligned b64 fragment loads, 16 distinct banks across lanes

#define WMMA_F32(A, B, C) \
  __builtin_amdgcn_wmma_f32_16x16x4_f32(false, (A), false, (B), (short)0, (C), false, false)

// ---------- init keys to +inf ----------
__global__ __launch_bounds__(256) void k_init_keys(unsigned long long* keys) {
  int t = blockIdx.x * 256 + threadIdx.x;
  if (t < TP) keys[t] = ~0ull;
}

// ---------- codebook squared norms (one wave32 per bin) ----------
__global__ __launch_bounds__(256) void k_cnorm(const float* __restrict__ cb,
                                               float* __restrict__ cnorm) {
  int wv = threadIdx.x >> 5;
  int lane = threadIdx.x & 31;
  int row = blockIdx.x * 8 + wv;
  const float* p = cb + (size_t)row * DD;
  float s = 0.f;
  for (int i = lane; i < DD; i += 32) { float v = p[i]; s += v * v; }
  for (int m = 16; m; m >>= 1) s += __shfl_xor(s, m, 32);
  if (lane == 0) cnorm[row] = s;
}

// ---------- conv-as-GEMM: Y[8192,768] = A[8192,1536] * W2[1536,768] + bias ----------
// A[t', 2*di+p] = x0[di*TT + 2*t' + p];  W2[j, n] = conv_w[n*K1 + j]
// Block tile 64(M) x 128(N); 8 waves in 2x4; each wave: 2 M-tiles x 2 N-tiles.
// LDS layout [row][k] so each v2f fragment (k,k+1) is one aligned ds_load_b64.
__global__ __launch_bounds__(256) void k_conv(const float* __restrict__ x0,
                                              const float* __restrict__ w,
                                              const float* __restrict__ bias,
                                              float* __restrict__ Y) {
  __shared__ float As[64][LKP];    // [t][kk]
  __shared__ float Bs[128][LKP];   // [n][kk]
  const int tb = blockIdx.x * 64;
  const int nb = blockIdx.y * 128;
  const int tid  = threadIdx.x;
  const int lane = tid & 31;
  const int wv   = tid >> 5;
  const int mr   = wv & 1;            // M pair (rows mr*32 .. +31)
  const int nc   = wv >> 1;           // N quad (cols nc*32 .. +31)
  const int m    = lane & 15;
  const int ksel = (lane >> 4) << 1;  // 0 or 2 (16x16x4 f32 half-wave K split)

  v8f acc00 = {}, acc01 = {}, acc10 = {}, acc11 = {};

  for (int jc = 0; jc < K1; jc += 16) {
    // Stage A: 8 d_in rows, 128 contiguous floats each (64 t' x 2 taps)
    {
      int dl = tid >> 5;          // 0..7
      int q  = tid & 31;          // 0..31
      const float4 v = *(const float4*)(x0 + (size_t)((jc >> 1) + dl) * TT + 2 * tb + 4 * q);
      int jj = 2 * dl;
      v2f lo; lo.x = v.x; lo.y = v.y;   // (t'=2q,   j=jj..jj+1)
      v2f hi; hi.x = v.z; hi.y = v.w;   // (t'=2q+1, j=jj..jj+1)
      *(v2f*)&As[2 * q + 0][jj] = lo;
      *(v2f*)&As[2 * q + 1][jj] = hi;
    }
    // Stage B: 128 output-channel rows, contiguous in j; one b128 store each
#pragma unroll
    for (int s = 0; s < 2; ++s) {
      int idx = tid + 256 * s;
      int nn = idx >> 2;          // 0..127
      int q  = idx & 3;           // 0..3
      const float4 v = *(const float4*)(w + (size_t)(nb + nn) * K1 + jc + 4 * q);
      *(float4*)&Bs[nn][4 * q] = v;
    }
    __syncthreads();
#pragma unroll
    for (int k0 = 0; k0 < 16; k0 += 4) {
      const int kk = k0 + ksel;
      v2f a0 = *(const v2f*)&As[mr * 32 + m][kk];
      v2f a1 = *(const v2f*)&As[mr * 32 + 16 + m][kk];
      v2f b0 = *(const v2f*)&Bs[nc * 32 + m][kk];
      v2f b1 = *(const v2f*)&Bs[nc * 32 + 16 + m][kk];
      acc00 = WMMA_F32(a0, b0, acc00);
      acc01 = WMMA_F32(a0, b1, acc01);
      acc10 = WMMA_F32(a1, b0, acc10);
      acc11 = WMMA_F32(a1, b1, acc11);
    }
    __syncthreads();
  }

  const int col0 = nb + nc * 32 + m;
  const int col1 = col0 + 16;
  const float cb0 = bias[col0], cb1 = bias[col1];
  const int r0 = tb + mr * 32 + ((lane >> 4) << 3);
  const int r1 = r0 + 16;
#pragma unroll
  for (int e = 0; e < 8; ++e) {
    Y[(size_t)(r0 + e) * DD + col0] = acc00[e] + cb0;
    Y[(size_t)(r0 + e) * DD + col1] = acc01[e] + cb1;
    Y[(size_t)(r1 + e) * DD + col0] = acc10[e] + cb0;
    Y[(size_t)(r1 + e) * DD + col1] = acc11[e] + cb1;
  }
}

__device__ inline unsigned int ford(float f) {
  unsigned int u = __float_as_uint(f);
  return (u & 0x80000000u) ? ~u : (u | 0x80000000u);  // order-preserving for min
}

// ---------- scores GEMM + argmin:  min_k (cnorm[k] - 2 * Y.row . codebook[k]) ----------
__global__ __launch_bounds__(256) void k_argmin(const float* __restrict__ Y,
                                                const float* __restrict__ cb,
                                                const float* __restrict__ cnorm,
                                                unsigned long long* __restrict__ keys) {
  __shared__ float As[64][LKP];
  __shared__ float Bs[128][LKP];
  const int tb = blockIdx.x * 64;
  const int nb = blockIdx.y * 128;
  const int tid  = threadIdx.x;
  const int lane = tid & 31;
  const int wv   = tid >> 5;
  const int mr   = wv & 1;
  const int nc   = wv >> 1;
  const int m    = lane & 15;
  const int ksel = (lane >> 4) << 1;

  v8f acc00 = {}, acc01 = {}, acc10 = {}, acc11 = {};

  for (int kc = 0; kc < DD; kc += 16) {
    {
      int tt = tid >> 2;  int q = tid & 3;
      const float4 v = *(const float4*)(Y + (size_t)(tb + tt) * DD + kc + 4 * q);
      *(float4*)&As[tt][4 * q] = v;
    }
#pragma unroll
    for (int s = 0; s < 2; ++s) {
      int idx = tid + 256 * s;
      int nn = idx >> 2;  int q = idx & 3;
      const float4 v = *(const float4*)(cb + (size_t)(nb + nn) * DD + kc + 4 * q);
      *(float4*)&Bs[nn][4 * q] = v;
    }
    __syncthreads();
#pragma unroll
    for (int k0 = 0; k0 < 16; k0 += 4) {
      const int kk = k0 + ksel;
      v2f a0 = *(const v2f*)&As[mr * 32 + m][kk];
      v2f a1 = *(const v2f*)&As[mr * 32 + 16 + m][kk];
      v2f b0 = *(const v2f*)&Bs[nc * 32 + m][kk];
      v2f b1 = *(const v2f*)&Bs[nc * 32 + 16 + m][kk];
      acc00 = WMMA_F32(a0, b0, acc00);
      acc01 = WMMA_F32(a0, b1, acc01);
      acc10 = WMMA_F32(a1, b0, acc10);
      acc11 = WMMA_F32(a1, b1, acc11);
    }
    __syncthreads();
  }

  const int n0 = nb + nc * 32 + m;
  const int n1 = n0 + 16;
  const float cn0 = cnorm[n0], cn1 = cnorm[n1];
  const int r0 = tb + mr * 32 + ((lane >> 4) << 3);
  const int r1 = r0 + 16;

#pragma unroll
  for (int half = 0; half < 2; ++half) {
    const v8f& aL = half ? acc10 : acc00;
    const v8f& aR = half ? acc11 : acc01;
    const int rb = half ? r1 : r0;
#pragma unroll
    for (int e = 0; e < 8; ++e) {
      float v0 = cn0 - 2.0f * aL[e];
      float v1 = cn1 - 2.0f * aR[e];
      float v; int idx;
      if (v1 < v0) { v = v1; idx = n1; } else { v = v0; idx = n0; }
      // reduce across the 16 N-lanes; xor masks 1..8 stay within each half-wave,
      // matching the C/D layout (lanes 0-15: rows M=0..7, lanes 16-31: rows M=8..15)
      for (int mask = 1; mask < 16; mask <<= 1) {
        float ov = __shfl_xor(v, mask, 32);
        int   oi = __shfl_xor(idx, mask, 32);
        if (ov < v || (ov == v && oi < idx)) { v = ov; idx = oi; }
      }
      if (m == 0) {
        unsigned long long key = ((unsigned long long)ford(v) << 32) | (unsigned int)idx;
        atomicMin(&keys[rb + e], key);
      }
    }
  }
}

// ---------- finalize: emit codes (low 32 bits of winning key) ----------
__global__ __launch_bounds__(256) void k_final(const unsigned long long* __restrict__ keys,
                                               float* __restrict__ out) {
  int t = blockIdx.x * 256 + threadIdx.x;
  if (t < TP) out[t] = (float)(unsigned int)(keys[t] & 0xFFFFFFFFu);
}

extern "C" void kernel_launch(void* const* d_in, const int* in_sizes, int n_in,
                              void* d_out, int out_size, void* d_ws, size_t ws_size,
                              hipStream_t stream) {
  const float* ssl      = (const float*)d_in[0];  // [8,768,16384] -> batch 0 = first D*T
  const float* conv_w   = (const float*)d_in[1];  // [768,768,2] row-major == W2^T
  const float* conv_b   = (const float*)d_in[2];  // [768]
  const float* codebook = (const float*)d_in[3];  // [1024,768]

  const size_t YBYTES = (size_t)TP * DD * sizeof(float);
  float* Y = (float*)d_ws;
  unsigned long long* keys = (unsigned long long*)((char*)d_ws + YBYTES);
  float* cnorm = (float*)((char*)d_ws + YBYTES + (size_t)TP * sizeof(unsigned long long));

  k_init_keys<<<TP / 256, 256, 0, stream>>>(keys);
  k_cnorm<<<NBINS / 8, 256, 0, stream>>>(codebook, cnorm);
  k_conv<<<dim3(TP / 64, DD / 128), 256, 0, stream>>>(ssl, conv_w, conv_b, Y);
  k_argmin<<<dim3(TP / 64, NBINS / 128), 256, 0, stream>>>(Y, codebook, cnorm, keys);
  k_final<<<TP / 256, 256, 0, stream>>>(keys, (float*)d_out);
}